// MANNCell_90434831385057
// MI455X (gfx1250) — compile-verified
//
#include <hip/hip_runtime.h>
#include <math.h>

// ---- problem constants ----
#define LSTMN    512
#define MEMN     2048
#define MEMD     256
#define NREADS   4
#define EPSV     1e-8f
#define BT       64                 // batch == time axis of the scan
#define IN_RAW   256
#define IN_TOT   (IN_RAW + NREADS*MEMN)      // 8448
#define GATES    (4*LSTMN)                    // 2048
#define PDIM     ((2*MEMD+1)*NREADS)          // 2052
#define OUTW     (LSTMN + NREADS*MEMD)        // 1536

typedef float v2f __attribute__((ext_vector_type(2)));
typedef float v8f __attribute__((ext_vector_type(8)));

__device__ __forceinline__ float sigf(float x) { return 1.0f / (1.0f + expf(-x)); }

// ---------------------------------------------------------------------------
// agent-scope grid barrier (monotone counter, reset by k_init each launch)
// ---------------------------------------------------------------------------
__device__ __forceinline__ void grid_barrier(unsigned* bar, unsigned target, int tid) {
    __syncthreads();
    if (tid == 0) {
        __threadfence();  // make this block's global stores agent-visible
        __hip_atomic_fetch_add(bar, 1u, __ATOMIC_ACQ_REL, __HIP_MEMORY_SCOPE_AGENT);
        while (__hip_atomic_load(bar, __ATOMIC_ACQUIRE, __HIP_MEMORY_SCOPE_AGENT) < target)
            __builtin_amdgcn_s_sleep(1);
    }
    __syncthreads();
}

__global__ void k_init(unsigned* bar) { *bar = 0u; }

// ---------------------------------------------------------------------------
// GEMM1: xw[t][g] = sum_k ctrl[t][k] * W_ih[g][k] + (b_ih+b_hh)[g]
// ctrl = concat(x (64x256), wr (64x8192)).  f32 WMMA 16x16x4.
// block = 4 waves (one 16-row M tile each), grid.x = 128 N tiles.
// ---------------------------------------------------------------------------
__global__ __launch_bounds__(128)
void k_gemm_xw(const float* __restrict__ x, const float* __restrict__ wr,
               const float* __restrict__ Wih, const float* __restrict__ bih,
               const float* __restrict__ bhh, float* __restrict__ xw)
{
    const int lane = threadIdx.x & 31;
    const int wv   = threadIdx.x >> 5;     // M tile 0..3
    const int nt   = blockIdx.x;           // N tile 0..127
    const int half = lane >> 4;
    const int lm   = lane & 15;
    const int m    = wv * 16 + lm;         // A row (time index)
    const int n    = nt * 16 + lm;         // B col (gate row)
    const float* wrow = Wih + (size_t)n * IN_TOT;
    const float* xrow = x  + (size_t)m * IN_RAW;
    const float* rrow = wr + (size_t)m * (NREADS * MEMN);

    v8f acc = {0.f, 0.f, 0.f, 0.f, 0.f, 0.f, 0.f, 0.f};
    for (int k0 = 0; k0 < IN_TOT; k0 += 4) {
        const int ka = k0 + 2 * half;
        v2f a, b;
        a[0] = (ka     < IN_RAW) ? xrow[ka]     : rrow[ka     - IN_RAW];
        a[1] = (ka + 1 < IN_RAW) ? xrow[ka + 1] : rrow[ka + 1 - IN_RAW];
        b[0] = wrow[ka];
        b[1] = wrow[ka + 1];
        acc = __builtin_amdgcn_wmma_f32_16x16x4_f32(false, a, false, b,
                                                    (short)0, acc, false, false);
    }
    const float bias = bih[n] + bhh[n];
#pragma unroll
    for (int i = 0; i < 8; ++i) {
        const int dm = wv * 16 + half * 8 + i;          // D: VGPR i -> M=i / i+8
        xw[(size_t)dm * GATES + n] = acc[i] + bias;
    }
}

// ---------------------------------------------------------------------------
// Persistent sequential LSTM scan. 128 blocks x 128 threads, 4 hidden units
// per block; each block keeps its 16 W_hh rows (4 gates x 4 units) in LDS.
// ---------------------------------------------------------------------------
#define SBLK 128
#define STHR 128
#define UPB  4

__global__ __launch_bounds__(STHR)
void k_lstm_scan(const float* __restrict__ xw, const float* __restrict__ Whh,
                 const float* __restrict__ h0, const float* __restrict__ c0,
                 float* __restrict__ hbuf, float* __restrict__ hs,
                 float* __restrict__ out, unsigned* __restrict__ bar)
{
    __shared__ float Wl[16][520];     // padded stride: conflict-free strided dot
    __shared__ float hl[LSTMN];
    __shared__ float part[STHR];
    __shared__ float g[16];
    __shared__ float cl[UPB];
    const int tid = threadIdx.x, bId = blockIdx.x;

    for (int idx = tid; idx < 16 * LSTMN; idx += STHR) {
        const int r = idx >> 9, k = idx & 511;
        const int grow = (r >> 2) * LSTMN + bId * UPB + (r & 3);  // gate*512 + unit
        Wl[r][k] = Whh[(size_t)grow * LSTMN + k];
    }
    if (tid < UPB) {
        cl[tid] = c0[bId * UPB + tid];
        hbuf[bId * UPB + tid] = h0[bId * UPB + tid];   // h_buf parity 0 = h0
    }
    grid_barrier(bar, SBLK * 1u, tid);

    const int row = tid >> 3, pt = tid & 7;
    for (int t = 0; t < BT; ++t) {
        const int par = t & 1, nxt = par ^ 1;
        for (int idx = tid; idx < LSTMN; idx += STHR)
            hl[idx] = hbuf[par * LSTMN + idx];
        __syncthreads();
        float acc = 0.f;
#pragma unroll 8
        for (int j = 0; j < 64; ++j) {
            const int k = pt + 8 * j;
            acc += Wl[row][k] * hl[k];
        }
        part[tid] = acc;
        __syncthreads();
        if (tid < 16) {
            float s = 0.f;
#pragma unroll
            for (int p = 0; p < 8; ++p) s += part[tid * 8 + p];
            const int grow = (tid >> 2) * LSTMN + bId * UPB + (tid & 3);
            g[tid] = s + xw[(size_t)t * GATES + grow];
        }
        __syncthreads();
        if (tid < UPB) {
            const float gi = g[tid], gf = g[4 + tid], gc = g[8 + tid], go = g[12 + tid];
            const float c = sigf(gf) * cl[tid] + sigf(gi) * tanhf(gc);
            const float h = sigf(go) * tanhf(c);
            cl[tid] = c;
            const int gu = bId * UPB + tid;
            hbuf[nxt * LSTMN + gu] = h;
            hs[(size_t)t * LSTMN + gu] = h;
            out[(size_t)t * OUTW + gu] = h;     // out[:, :512] = hs
        }
        grid_barrier(bar, (unsigned)(SBLK * (t + 2)), tid);
    }
}

// ---------------------------------------------------------------------------
// GEMM2: param = hs @ W_p^T + b_p, fused tanh/sigmoid epilogue.
// N padded 2052 -> 129 tiles of 16 (guarded). f32 WMMA 16x16x4.
// ---------------------------------------------------------------------------
__global__ __launch_bounds__(128)
void k_gemm_param(const float* __restrict__ hs, const float* __restrict__ Wp,
                  const float* __restrict__ bp, float* __restrict__ kk,
                  float* __restrict__ aa, float* __restrict__ alpha)
{
    const int lane = threadIdx.x & 31;
    const int wv   = threadIdx.x >> 5;
    const int nt   = blockIdx.x;           // 0..128
    const int half = lane >> 4;
    const int lm   = lane & 15;
    const int m    = wv * 16 + lm;
    const int n    = nt * 16 + lm;
    const bool nv  = (n < PDIM);
    const float* wrow = Wp + (size_t)(nv ? n : 0) * LSTMN;
    const float* arow = hs + (size_t)m * LSTMN;

    v8f acc = {0.f, 0.f, 0.f, 0.f, 0.f, 0.f, 0.f, 0.f};
    for (int k0 = 0; k0 < LSTMN; k0 += 4) {
        const int ka = k0 + 2 * half;
        v2f a, b;
        a[0] = arow[ka];
        a[1] = arow[ka + 1];
        b[0] = nv ? wrow[ka]     : 0.f;    // cndmask select: EXEC stays all-1 for WMMA
        b[1] = nv ? wrow[ka + 1] : 0.f;
        acc = __builtin_amdgcn_wmma_f32_16x16x4_f32(false, a, false, b,
                                                    (short)0, acc, false, false);
    }
    if (!nv) return;
    const float bias = bp[n];
#pragma unroll
    for (int i = 0; i < 8; ++i) {
        const int t = wv * 16 + half * 8 + i;
        const float v = acc[i] + bias;
        if (n < NREADS * MEMD)
            kk[(size_t)t * (NREADS * MEMD) + n] = tanhf(v);
        else if (n < 2 * NREADS * MEMD)
            aa[(size_t)t * (NREADS * MEMD) + (n - NREADS * MEMD)] = tanhf(v);
        else
            alpha[t * NREADS + (n - 2 * NREADS * MEMD)] = sigf(v);
    }
}

// ---------------------------------------------------------------------------
// Bottom-4 of wu per row (4 masked argmin passes).  widx[b][0] = lu_idx.
// ---------------------------------------------------------------------------
__global__ __launch_bounds__(256)
void k_top4min(const float* __restrict__ wu, int* __restrict__ widx)
{
    __shared__ float sv[256];
    __shared__ int   si[256];
    __shared__ int   chosen[4];
    const int b = blockIdx.x, tid = threadIdx.x;
    const float* row = wu + (size_t)b * MEMN;
    for (int j = 0; j < NREADS; ++j) {
        float bv = INFINITY; int bi = MEMN;
        for (int n = tid; n < MEMN; n += 256) {
            bool skip = false;
            for (int q = 0; q < j; ++q) skip |= (n == chosen[q]);
            if (skip) continue;
            const float v = row[n];
            if (v < bv || (v == bv && n < bi)) { bv = v; bi = n; }
        }
        sv[tid] = bv; si[tid] = bi;
        __syncthreads();
        for (int s = 128; s > 0; s >>= 1) {
            if (tid < s) {
                const float ov = sv[tid + s]; const int oi = si[tid + s];
                if (ov < sv[tid] || (ov == sv[tid] && oi < si[tid])) { sv[tid] = ov; si[tid] = oi; }
            }
            __syncthreads();
        }
        if (tid == 0) { chosen[j] = si[0]; widx[b * NREADS + j] = si[0]; }
        __syncthreads();
    }
}

// ---------------------------------------------------------------------------
// Cosine similarity: single HBM pass over M (memory-bound -> VALU float4).
// cosw[b][r][n] = <k_r, M_n> / (||k_r|| * ||M_n|| + eps)
// ---------------------------------------------------------------------------
__global__ __launch_bounds__(256)
void k_cosine(const float* __restrict__ M, const float* __restrict__ kk,
              float* __restrict__ cosw)
{
    __shared__ float kl[NREADS][MEMD];
    __shared__ float kn[NREADS];
    const int b  = blockIdx.x >> 3;
    const int n0 = (blockIdx.x & 7) * 256;
    const int tid = threadIdx.x;
    for (int i = tid; i < NREADS * MEMD; i += 256)
        kl[i >> 8][i & 255] = kk[(size_t)b * (NREADS * MEMD) + i];
    __syncthreads();
    if (tid < NREADS) {
        float s = 0.f;
        for (int d = 0; d < MEMD; ++d) { const float v = kl[tid][d]; s += v * v; }
        kn[tid] = sqrtf(s);
    }
    __syncthreads();
    const int n = n0 + tid;
    const float4* mp = (const float4*)(M + ((size_t)b * MEMN + n) * MEMD);
    float msq = 0.f;
    float a[NREADS] = {0.f, 0.f, 0.f, 0.f};
    for (int q = 0; q < MEMD / 4; ++q) {
        const float4 m4 = mp[q];
        msq += m4.x * m4.x + m4.y * m4.y + m4.z * m4.z + m4.w * m4.w;
        const int d = 4 * q;
#pragma unroll
        for (int r = 0; r < NREADS; ++r)
            a[r] += kl[r][d] * m4.x + kl[r][d + 1] * m4.y + kl[r][d + 2] * m4.z + kl[r][d + 3] * m4.w;
    }
    const float mn = sqrtf(msq);
#pragma unroll
    for (int r = 0; r < NREADS; ++r)
        cosw[((size_t)b * NREADS + r) * MEMN + n] = a[r] / (kn[r] * mn + EPSV);
}

// ---------------------------------------------------------------------------
// Row softmax over 2048 (in place).  256 blocks = (b, r).
// ---------------------------------------------------------------------------
__global__ __launch_bounds__(256)
void k_softmax(float* __restrict__ w)
{
    __shared__ float red[256];
    const int tid = threadIdx.x;
    float* row = w + (size_t)blockIdx.x * MEMN;
    float lv[8];
    float mx = -INFINITY;
#pragma unroll
    for (int i = 0; i < 8; ++i) { lv[i] = row[tid + 256 * i]; mx = fmaxf(mx, lv[i]); }
    red[tid] = mx; __syncthreads();
    for (int s = 128; s > 0; s >>= 1) { if (tid < s) red[tid] = fmaxf(red[tid], red[tid + s]); __syncthreads(); }
    mx = red[0]; __syncthreads();
    float sum = 0.f;
#pragma unroll
    for (int i = 0; i < 8; ++i) { lv[i] = expf(lv[i] - mx); sum += lv[i]; }
    red[tid] = sum; __syncthreads();
    for (int s = 128; s > 0; s >>= 1) { if (tid < s) red[tid] += red[tid + s]; __syncthreads(); }
    const float inv = 1.f / red[0];
#pragma unroll
    for (int i = 0; i < 8; ++i) row[tid + 256 * i] = lv[i] * inv;
}

// ---------------------------------------------------------------------------
// G[b][r][r'] = sa[r'] * <wr_t[r], wr_prev[r']> + (1-sa[r']) * sum_q wr_t[r][widx[q]]
// (collapses wr_t @ ww_t^T so M_t is never materialized)
// ---------------------------------------------------------------------------
__global__ __launch_bounds__(256)
void k_gmat(const float* __restrict__ wrt, const float* __restrict__ wrp,
            const int* __restrict__ widx, const float* __restrict__ alpha,
            float* __restrict__ G)
{
    __shared__ float red[256];
    __shared__ float S[16];
    __shared__ float tsum[NREADS];
    const int b = blockIdx.x, tid = threadIdx.x;
    const float* wt = wrt + (size_t)b * NREADS * MEMN;
    const float* wp = wrp + (size_t)b * NREADS * MEMN;
    float s[16];
#pragma unroll
    for (int p = 0; p < 16; ++p) s[p] = 0.f;
    for (int n = tid; n < MEMN; n += 256) {
        float tv[NREADS], pv[NREADS];
#pragma unroll
        for (int r = 0; r < NREADS; ++r) { tv[r] = wt[r * MEMN + n]; pv[r] = wp[r * MEMN + n]; }
#pragma unroll
        for (int r = 0; r < NREADS; ++r)
#pragma unroll
            for (int q = 0; q < NREADS; ++q) s[r * 4 + q] += tv[r] * pv[q];
    }
    for (int p = 0; p < 16; ++p) {
        red[tid] = s[p]; __syncthreads();
        for (int st = 128; st > 0; st >>= 1) { if (tid < st) red[tid] += red[tid + st]; __syncthreads(); }
        if (tid == 0) S[p] = red[0];
        __syncthreads();
    }
    if (tid < NREADS) {
        float ts = 0.f;
        for (int q = 0; q < NREADS; ++q) ts += wt[tid * MEMN + widx[b * NREADS + q]];
        tsum[tid] = ts;
    }
    __syncthreads();
    if (tid < 16) {
        const int r = tid >> 2, rp = tid & 3;
        const float sa = alpha[b * NREADS + rp];
        G[b * 16 + tid] = sa * S[r * 4 + rp] + (1.f - sa) * tsum[r];
    }
}

// ---------------------------------------------------------------------------
// r_t[b][r][d] = sum_n wr_t[r][n]*erase[n]*M[b][n][d] + sum_r' G[r][r']*a[r'][d]
// second pass over M served from L2 (134 MB < 192 MB).
// ---------------------------------------------------------------------------
__global__ __launch_bounds__(256)
void k_readout(const float* __restrict__ M, const float* __restrict__ wrt,
               const float* __restrict__ aa, const float* __restrict__ G,
               const int* __restrict__ widx, float* __restrict__ out)
{
    __shared__ float wl[NREADS][256];
    const int b = blockIdx.x, d = threadIdx.x;
    const int lu = widx[b * NREADS + 0];
    const float* wt = wrt + (size_t)b * NREADS * MEMN;
    const float* Mb = M + (size_t)b * MEMN * MEMD;
    float acc[NREADS] = {0.f, 0.f, 0.f, 0.f};
    for (int c = 0; c < MEMN / 256; ++c) {
        const int n = c * 256 + d;
        const float er = (n == lu) ? 0.f : 1.f;
#pragma unroll
        for (int r = 0; r < NREADS; ++r) wl[r][d] = wt[r * MEMN + n] * er;
        __syncthreads();
        for (int nn = 0; nn < 256; ++nn) {
            const float m = Mb[(size_t)(c * 256 + nn) * MEMD + d];
#pragma unroll
            for (int r = 0; r < NREADS; ++r) acc[r] += wl[r][nn] * m;
        }
        __syncthreads();
    }
#pragma unroll
    for (int r = 0; r < NREADS; ++r) {
        float v = acc[r];
#pragma unroll
        for (int rp = 0; rp < NREADS; ++rp)
            v += G[b * 16 + r * 4 + rp] * aa[(size_t)b * (NREADS * MEMD) + rp * MEMD + d];
        out[(size_t)b * OUTW + LSTMN + r * MEMD + d] = v;
    }
}

// ---------------------------------------------------------------------------
extern "C" void kernel_launch(void* const* d_in, const int* in_sizes, int n_in,
                              void* d_out, int out_size, void* d_ws, size_t ws_size,
                              hipStream_t stream)
{
    (void)in_sizes; (void)n_in; (void)out_size; (void)ws_size;
    const float* x    = (const float*)d_in[0];
    const float* M    = (const float*)d_in[1];
    const float* wu   = (const float*)d_in[2];
    const float* wr   = (const float*)d_in[3];
    const float* h0   = (const float*)d_in[4];
    const float* c0   = (const float*)d_in[5];
    const float* Wih  = (const float*)d_in[6];
    const float* Whh  = (const float*)d_in[7];
    const float* bih  = (const float*)d_in[8];
    const float* bhh  = (const float*)d_in[9];
    const float* Wp   = (const float*)d_in[10];
    const float* bp   = (const float*)d_in[11];
    float* out = (float*)d_out;

    // workspace layout (floats), total ~3.2 MB
    float* f     = (float*)d_ws;
    float* xw    = f;                    // 64*2048
    float* hs    = xw    + 131072;       // 64*512
    float* hbuf  = hs    + 32768;        // 2*512
    float* kk    = hbuf  + 1024;         // 64*4*256
    float* aa    = kk    + 65536;        // 64*4*256
    float* alpha = aa    + 65536;        // 64*4
    float* cosw  = alpha + 256;          // 64*4*2048 (becomes wr_t after softmax)
    float* G     = cosw  + 524288;       // 64*16
    int*   widx  = (int*)(G + 1024);     // 64*4
    unsigned* bar = (unsigned*)(widx + 256);

    k_init      <<<1,   1,   0, stream>>>(bar);
    k_gemm_xw   <<<128, 128, 0, stream>>>(x, wr, Wih, bih, bhh, xw);
    k_lstm_scan <<<SBLK, STHR, 0, stream>>>(xw, Whh, h0, c0, hbuf, hs, out, bar);
    k_gemm_param<<<129, 128, 0, stream>>>(hs, Wp, bp, kk, aa, alpha);
    k_top4min   <<<64,  256, 0, stream>>>(wu, widx);
    k_cosine    <<<512, 256, 0, stream>>>(M, kk, cosw);
    k_softmax   <<<256, 256, 0, stream>>>(cosw);
    k_gmat      <<<64,  256, 0, stream>>>(cosw, wr, widx, alpha, G);
    k_readout   <<<64,  256, 0, stream>>>(M, cosw, aa, G, widx, out);
}